// sparseFeaturePnP_67637144978499
// MI455X (gfx1250) — compile-verified
//
#include <hip/hip_runtime.h>
#include <math.h>

// ---------------------------------------------------------------------------
// Sparse feature-metric PnP (Gauss-Newton, 5 iters) for MI455X / gfx1250.
// Memory-bound gather + f32 WMMA (16x16x4) reduction of H = J^T J, g = J^T e.
// Row layout per (n,c): [J0..J5, e, 0 x 9]  ->  D += Rw^T Rw accumulates
//   H[0..5][0..5], g = D[0..5][6] in one chained WMMA accumulator per wave.
// ---------------------------------------------------------------------------

typedef float v2f __attribute__((ext_vector_type(2)));
typedef float v8f __attribute__((ext_vector_type(8)));

#define N_ITERS     5
#define LAMBDA_LM   0.01f

#define NBLOCKS_ACC 256
#define THREADS_ACC 256
#define WAVES_PER_BLOCK (THREADS_ACC / 32)
#define TOTAL_WAVES (NBLOCKS_ACC * WAVES_PER_BLOCK)

// workspace layout (floats):
//   [0..8]  R state   [9..11] t state   [12..15] pad
//   [16 .. 16 + NBLOCKS_ACC*49)  per-block partial H7 (7x7 row-major)
#define WS_STATE_F   0
#define WS_PART_F    16

// ---------------------------------------------------------------------------
__global__ void pnp_init_kernel(const float* __restrict__ R0,
                                const float* __restrict__ t0,
                                float* __restrict__ state) {
  int i = threadIdx.x;
  if (i < 9)       state[i] = R0[i];
  else if (i < 12) state[i] = t0[i - 9];
  else if (i < 16) state[i] = 0.0f;
}

// ---------------------------------------------------------------------------
// Accumulate per-block partial 7x7 (H | g) via WMMA f32 16x16x4.
// Assumes C is a multiple of 32 (C=128 here).
// ---------------------------------------------------------------------------
__global__ __launch_bounds__(THREADS_ACC)
void pnp_accum_kernel(const float* __restrict__ pts3D,   // N x 3
                      const float* __restrict__ fref,    // N x C
                      const float* __restrict__ fmap,    // C x fmH x fmW
                      const float* __restrict__ gxmap,   // C x fmH x fmW
                      const float* __restrict__ gymap,   // C x fmH x fmW
                      const float* __restrict__ Kmat,    // 3 x 3
                      const float* __restrict__ state,   // R(9), t(3)
                      float* __restrict__ partials,      // NBLOCKS_ACC x 49
                      const int* __restrict__ imw_p,
                      const int* __restrict__ imh_p,
                      int N, int C) {
  __shared__ float ldsRows[WAVES_PER_BLOCK][32 * 16]; // per-wave 32 rows x 16
  __shared__ float ldsHg[49];

  const int tid  = threadIdx.x;
  const int lane = tid & 31;
  const int wave = tid >> 5;

  // zero the never-rewritten row columns 7..15 and the block accumulator
  float* myrow = &ldsRows[wave][lane * 16];
#pragma unroll
  for (int k = 7; k < 16; ++k) myrow[k] = 0.0f;
  if (tid < 49) ldsHg[tid] = 0.0f;
  __syncthreads();

  const int imW = imw_p[0];
  const int imH = imh_p[0];
  const int fmW = imW;           // reference setup: map size == image size
  const int fmH = imH;
  const int planeSz = fmH * fmW;

  const float R00 = state[0], R01 = state[1], R02 = state[2];
  const float R10 = state[3], R11 = state[4], R12 = state[5];
  const float R20 = state[6], R21 = state[7], R22 = state[8];
  const float tx = state[9], ty = state[10], tz = state[11];
  const float fx = Kmat[0], cx = Kmat[2];
  const float fy = Kmat[4], cy = Kmat[5];

  v8f acc = {0.f, 0.f, 0.f, 0.f, 0.f, 0.f, 0.f, 0.f};

  const int waveGlobal = blockIdx.x * WAVES_PER_BLOCK + wave;
  const int colIdx = lane & 15;
  const int half   = (lane >> 4) << 1;  // 0 for lanes 0-15, 2 for lanes 16-31
  float* ldsw = &ldsRows[wave][0];

  for (int n = waveGlobal; n < N; n += TOTAL_WAVES) {
    // ---- per-point geometry (uniform across the wave; redundant per lane) --
    const float X = pts3D[3 * n + 0];
    const float Y = pts3D[3 * n + 1];
    const float Z = pts3D[3 * n + 2];
    const float x = R00 * X + R01 * Y + R02 * Z + tx;
    const float y = R10 * X + R11 * Y + R12 * Z + ty;
    const float z = R20 * X + R21 * Y + R22 * Z + tz;
    const float u = fx * x + cx * z;
    const float v = fy * y + cy * z;
    const float invw = 1.0f / z;
    const int px = (int)rintf(u * invw) - 1;
    const int py = (int)rintf(v * invw) - 1;
    const float mask =
        (px >= 0 && py >= 0 && px < imW && py < imH) ? 1.0f : 0.0f;
    int row = (int)floorf((float)py * (float)fmH / (float)imH);
    row = min(max(row, 0), fmH - 1);
    int col = (int)floorf((float)px * (float)fmW / (float)imW);
    col = min(max(col, 0), fmW - 1);

    const float zinv = 1.0f / z;
    const float a  = fx * zinv;
    const float b  = fy * zinv;
    const float cJ = -fx * x * zinv * zinv;
    const float d  = -fy * y * zinv * zinv;
    // A = J_px_p @ [I3 | -skew(p3d)]  (2 x 6), premultiplied by mask
    float A0[6] = { a, 0.0f, cJ, cJ * y, a * z - cJ * x, -a * y };
    float A1[6] = { 0.0f, b, d, -b * z + d * y, -d * x, b * x };
#pragma unroll
    for (int k = 0; k < 6; ++k) { A0[k] *= mask; A1[k] *= mask; }

    const int gbase = row * fmW + col;
    const float* fm = fmap  + gbase;
    const float* gm = gxmap + gbase;
    const float* hm = gymap + gbase;
    const long frefBase = (long)n * C;

    // ---- channel loop: 32 channels per chunk, one per lane -----------------
    for (int chunk = 0; chunk < C; chunk += 32) {
      const int c = chunk + lane;
      const float f  = fm[c * planeSz];
      const float gx = gm[c * planeSz];
      const float gy = hm[c * planeSz];
      const float e  = f - fref[frefBase + c];

      myrow[0] = gx * A0[0] + gy * A1[0];
      myrow[1] = gx * A0[1] + gy * A1[1];
      myrow[2] = gx * A0[2] + gy * A1[2];
      myrow[3] = gx * A0[3] + gy * A1[3];
      myrow[4] = gx * A0[4] + gy * A1[4];
      myrow[5] = gx * A0[5] + gy * A1[5];
      myrow[6] = e;

      // make this wave's LDS stores visible before cross-lane re-reads
      asm volatile("s_wait_dscnt 0" ::: "memory");

      // 8 WMMAs consume 32 rows, 4 per instruction (K=4).
      // Operand layout (16x4 f32 A): lane m<16 holds {row r0, row r0+1}[m],
      // lane m+16 holds {row r0+2, row r0+3}[m] -> identical registers serve
      // as both A (=Rw^T) and B (=Rw), so D += Rw^T Rw.
#pragma unroll
      for (int q = 0; q < 8; ++q) {
        const int r0 = q * 4 + half;
        v2f av;
        av.x = ldsw[(r0 + 0) * 16 + colIdx];
        av.y = ldsw[(r0 + 1) * 16 + colIdx];
        acc = __builtin_amdgcn_wmma_f32_16x16x4_f32(
            /*neg_a=*/false, av, /*neg_b=*/false, av,
            /*c_mod=*/(short)0, acc, /*reuse_a=*/false, /*reuse_b=*/false);
      }
      // ensure LDS reads retired before next chunk overwrites rows
      asm volatile("s_wait_dscnt 0" ::: "memory");
    }
  }

  // ---- combine the 8 wave accumulators into the block partial --------------
  // D element (M = vgpr + 8*(lane>=16), Ncol = lane%16); we need M,N in 0..6,
  // i.e. lanes 0..6, vgprs 0..6.
  if (lane < 7) {
#pragma unroll
    for (int vv = 0; vv < 7; ++vv) {
      atomicAdd(&ldsHg[vv * 7 + lane], acc[vv]);
    }
  }
  __syncthreads();
  if (tid < 49) partials[blockIdx.x * 49 + tid] = ldsHg[tid];
}

// ---------------------------------------------------------------------------
// Reduce partials, LM-damp, solve 6x6, so3exp, update (R,t) state + d_out.
// ---------------------------------------------------------------------------
__global__ void pnp_solve_kernel(const float* __restrict__ partials,
                                 float* __restrict__ state,
                                 float* __restrict__ out) {
  __shared__ float Hg[49];
  const int tid = threadIdx.x;
  for (int j = tid; j < 49; j += blockDim.x) {
    float s = 0.0f;
    for (int b = 0; b < NBLOCKS_ACC; ++b) s += partials[b * 49 + j];
    Hg[j] = s;
  }
  __syncthreads();

  if (tid == 0) {
    float M[6][7];
    for (int i = 0; i < 6; ++i) {
      for (int j = 0; j < 6; ++j) M[i][j] = Hg[i * 7 + j];
      M[i][i] += LAMBDA_LM * (M[i][i] + 1e-9f);
      M[i][6] = -Hg[i * 7 + 6];  // -g
    }
    // Gaussian elimination with partial pivoting
    for (int k = 0; k < 6; ++k) {
      int piv = k;
      float mx = fabsf(M[k][k]);
      for (int r = k + 1; r < 6; ++r) {
        float arr = fabsf(M[r][k]);
        if (arr > mx) { mx = arr; piv = r; }
      }
      if (piv != k) {
        for (int j = k; j < 7; ++j) {
          float tmp = M[k][j]; M[k][j] = M[piv][j]; M[piv][j] = tmp;
        }
      }
      const float inv = 1.0f / M[k][k];
      for (int r = k + 1; r < 6; ++r) {
        const float fac = M[r][k] * inv;
        for (int j = k; j < 7; ++j) M[r][j] -= fac * M[k][j];
      }
    }
    float delta[6];
    for (int i = 5; i >= 0; --i) {
      float s = M[i][6];
      for (int j = i + 1; j < 6; ++j) s -= M[i][j] * delta[j];
      delta[i] = s / M[i][i];
    }

    // dR = so3exp(delta[3:6])
    const float wx = delta[3], wy = delta[4], wz = delta[5];
    const float th = sqrtf(wx * wx + wy * wy + wz * wz + 1e-24f);
    const float Af = sinf(th) / th;
    const float Bf = (1.0f - cosf(th)) / (th * th);
    // W = skew(w)
    const float W[3][3] = { {0.f, -wz,  wy},
                            { wz, 0.f, -wx},
                            {-wy,  wx, 0.f} };
    float W2[3][3];
    for (int i = 0; i < 3; ++i)
      for (int j = 0; j < 3; ++j) {
        float s = 0.f;
        for (int k = 0; k < 3; ++k) s += W[i][k] * W[k][j];
        W2[i][j] = s;
      }
    float dR[3][3];
    for (int i = 0; i < 3; ++i)
      for (int j = 0; j < 3; ++j)
        dR[i][j] = (i == j ? 1.0f : 0.0f) + Af * W[i][j] + Bf * W2[i][j];

    // R = dR @ R ; t = dR @ t + delta[:3]
    float Rold[3][3], told[3];
    for (int i = 0; i < 3; ++i) {
      told[i] = state[9 + i];
      for (int j = 0; j < 3; ++j) Rold[i][j] = state[3 * i + j];
    }
    float Rnew[3][3], tnew[3];
    for (int i = 0; i < 3; ++i) {
      float ts = 0.f;
      for (int j = 0; j < 3; ++j) {
        float s = 0.f;
        for (int k = 0; k < 3; ++k) s += dR[i][k] * Rold[k][j];
        Rnew[i][j] = s;
        ts += dR[i][j] * told[j];
      }
      tnew[i] = ts + delta[i];
    }
    for (int i = 0; i < 3; ++i) {
      state[9 + i] = tnew[i];
      out[9 + i]   = tnew[i];
      for (int j = 0; j < 3; ++j) {
        state[3 * i + j] = Rnew[i][j];
        out[3 * i + j]   = Rnew[i][j];
      }
    }
  }
}

// ---------------------------------------------------------------------------
extern "C" void kernel_launch(void* const* d_in, const int* in_sizes, int n_in,
                              void* d_out, int out_size, void* d_ws,
                              size_t ws_size, hipStream_t stream) {
  const float* pts3D = (const float*)d_in[0];
  const float* fref  = (const float*)d_in[1];
  const float* fmap  = (const float*)d_in[2];
  const float* gxmap = (const float*)d_in[3];
  const float* gymap = (const float*)d_in[4];
  const float* Kmat  = (const float*)d_in[5];
  const float* R0    = (const float*)d_in[6];
  const float* t0    = (const float*)d_in[7];
  const int*   imw   = (const int*)d_in[8];
  const int*   imh   = (const int*)d_in[9];

  const int N = in_sizes[0] / 3;
  const int C = in_sizes[1] / N;

  float* state    = (float*)d_ws + WS_STATE_F;
  float* partials = (float*)d_ws + WS_PART_F;
  float* out      = (float*)d_out;

  pnp_init_kernel<<<1, 32, 0, stream>>>(R0, t0, state);
  for (int it = 0; it < N_ITERS; ++it) {
    pnp_accum_kernel<<<NBLOCKS_ACC, THREADS_ACC, 0, stream>>>(
        pts3D, fref, fmap, gxmap, gymap, Kmat, state, partials, imw, imh, N, C);
    pnp_solve_kernel<<<1, 64, 0, stream>>>(partials, state, out);
  }
}